// STSGT_Layer_28114855919851
// MI455X (gfx1250) — compile-verified
//
#include <hip/hip_runtime.h>

// ---------------------------------------------------------------------------
// STSGT layer for MI455X (gfx1250, wave32).
// Dense projections use V_WMMA_F32_16X16X4_F32 (full fp32 precision; GEMMs are
// ~32 GFLOP total and far below the WMMA roof, so no precision sacrifice).
// The K-loop issues clause'd fragment loads with staggered s_wait_loadcnt so
// WMMAs overlap in-flight loads; the streamed A tile is warmed up front with
// global_prefetch_b8. Edge softmax-aggregation uses L2-side f32 atomics
// (memory-bound stage, ~2-3 GB of mostly-L2 traffic -> the real roofline).
// ---------------------------------------------------------------------------

#define N_NODES 10000
#define T_STEPS 8
#define F_DIM   128
#define H_HEADS 4
#define C_DIM   32
#define E_EDGES 160000
#define M_ROWS  (N_NODES * T_STEPS)          // 80000 rows, = 5000 * 16
#define SCALE   0.17677669529663687f         // 1/sqrt(32)

typedef __attribute__((ext_vector_type(2))) float v2f;
typedef __attribute__((ext_vector_type(8))) float v8f;

#define WMMA_F32(a, b, c) \
    __builtin_amdgcn_wmma_f32_16x16x4_f32(false, (a), false, (b), (short)0, (c), false, false)

// ---------------------------------------------------------------------------
// WMMA fp32 GEMM:  Y[M,Nout] = A[M,K] @ W[Nout,K]^T + bias (+resid) (relu?)
// One wave computes a 16x64 tile: A fragment loaded once per K-step, reused by
// 4 WMMA accumulators. Layouts per CDNA5 ISA 7.12.2:
//   A 16x4 : lane(m=lane%16, khalf=lane/16), vgpr j -> k = 2*khalf + j
//   B 4x16 : lane(n=lane%16, khalf=lane/16), vgpr j -> k = 2*khalf + j
//   C/D    : vgpr r -> row m0 + r + 8*(lane/16), col n0 + lane%16
// ---------------------------------------------------------------------------
__global__ __launch_bounds__(256)
void gemm_wmma_f32(const float* __restrict__ A, const float* __restrict__ W,
                   const float* __restrict__ bias, const float* __restrict__ resid,
                   float* __restrict__ Y, int Mrows, int Nout, int K, int relu)
{
    const int lane = threadIdx.x & 31;
    const int wave = blockIdx.x * (blockDim.x >> 5) + (threadIdx.x >> 5);
    const int mTiles = Mrows >> 4;
    const int nGroups = Nout >> 6;
    if (wave >= mTiles * nGroups) return;
    // consecutive waves share one 64xK W tile (stays hot in cache); A streams.
    const int mt = wave % mTiles;
    const int ng = wave / mTiles;
    const int m0 = mt << 4, n0 = ng << 6;
    const int r16 = lane & 15;
    const int kh  = (lane >> 4) << 1;          // 0 or 2

    const float* arow = A + (size_t)(m0 + r16) * K + kh;
    const float* w0   = W + (size_t)(n0 + r16) * K + kh;
    const float* w1   = w0 + (size_t)16 * K;
    const float* w2   = w0 + (size_t)32 * K;
    const float* w3   = w0 + (size_t)48 * K;

    // Warm the streamed A tile: each lane prefetches its full K-row (512B for
    // K=128 -> 4 cachelines) so in-loop demand loads hit. Emits
    // global_prefetch_b8; no counters, fire-and-forget.
    for (int pb = 0; pb < K; pb += 32)
        __builtin_prefetch((const void*)(arow + pb), 0, 0);

    v8f acc0 = {}, acc1 = {}, acc2 = {}, acc3 = {};

    // software pipeline: fragments for step k are loaded during step k-4's math
    v2f a  = *(const v2f*)(arow);
    v2f b0 = *(const v2f*)(w0);
    v2f b1 = *(const v2f*)(w1);
    v2f b2 = *(const v2f*)(w2);
    v2f b3 = *(const v2f*)(w3);
    for (int k = 4; k < K; k += 4) {
        v2f an  = *(const v2f*)(arow + k);
        v2f b0n = *(const v2f*)(w0 + k);
        v2f b1n = *(const v2f*)(w1 + k);
        v2f b2n = *(const v2f*)(w2 + k);
        v2f b3n = *(const v2f*)(w3 + k);
        acc0 = WMMA_F32(a, b0, acc0);
        acc1 = WMMA_F32(a, b1, acc1);
        acc2 = WMMA_F32(a, b2, acc2);
        acc3 = WMMA_F32(a, b3, acc3);
        a = an; b0 = b0n; b1 = b1n; b2 = b2n; b3 = b3n;
    }
    acc0 = WMMA_F32(a, b0, acc0);
    acc1 = WMMA_F32(a, b1, acc1);
    acc2 = WMMA_F32(a, b2, acc2);
    acc3 = WMMA_F32(a, b3, acc3);

    const int rowBase = m0 + ((lane >> 4) << 3);   // +8 for upper half-wave
    const int col = n0 + r16;
    const float bb0 = bias[col], bb1 = bias[col + 16];
    const float bb2 = bias[col + 32], bb3 = bias[col + 48];
#pragma unroll
    for (int r = 0; r < 8; ++r) {
        const size_t ro = (size_t)(rowBase + r) * Nout;
        float v0 = acc0[r] + bb0;
        float v1 = acc1[r] + bb1;
        float v2v = acc2[r] + bb2;
        float v3 = acc3[r] + bb3;
        if (relu) {
            v0 = fmaxf(v0, 0.f); v1 = fmaxf(v1, 0.f);
            v2v = fmaxf(v2v, 0.f); v3 = fmaxf(v3, 0.f);
        }
        if (resid) {
            v0  += resid[ro + col];      v1 += resid[ro + col + 16];
            v2v += resid[ro + col + 32]; v3 += resid[ro + col + 48];
        }
        Y[ro + col]      = v0;  Y[ro + col + 16] = v1;
        Y[ro + col + 32] = v2v; Y[ro + col + 48] = v3;
    }
}

// ---------------------------------------------------------------------------
// Temporal self-attention: one thread per (node, head, t). T=8 fits registers.
// qkv layout: row = n*T + t, cols [0,128)=q, [128,256)=k, [256,384)=v.
// ---------------------------------------------------------------------------
__global__ __launch_bounds__(256)
void temporal_attn(const float* __restrict__ qkv, float* __restrict__ o)
{
    int idx = blockIdx.x * blockDim.x + threadIdx.x;
    if (idx >= N_NODES * H_HEADS * T_STEPS) return;
    const int t = idx % T_STEPS;
    const int h = (idx / T_STEPS) % H_HEADS;
    const int n = idx / (T_STEPS * H_HEADS);

    const float* base = qkv + (size_t)n * T_STEPS * 384;
    const float4* q4 = (const float4*)(base + t * 384 + h * C_DIM);
    float4 q[8];
#pragma unroll
    for (int i = 0; i < 8; ++i) q[i] = q4[i];

    float p[T_STEPS];
    float mx = -1e30f;
#pragma unroll
    for (int s = 0; s < T_STEPS; ++s) {
        const float4* k4 = (const float4*)(base + s * 384 + 128 + h * C_DIM);
        float acc = 0.f;
#pragma unroll
        for (int i = 0; i < 8; ++i) {
            float4 kv = k4[i];
            acc += q[i].x * kv.x + q[i].y * kv.y + q[i].z * kv.z + q[i].w * kv.w;
        }
        p[s] = acc * SCALE;
        mx = fmaxf(mx, p[s]);
    }
    float den = 0.f;
#pragma unroll
    for (int s = 0; s < T_STEPS; ++s) { p[s] = __expf(p[s] - mx); den += p[s]; }
    const float inv = 1.f / den;

    float4 out[8];
#pragma unroll
    for (int i = 0; i < 8; ++i) { out[i].x = out[i].y = out[i].z = out[i].w = 0.f; }
#pragma unroll
    for (int s = 0; s < T_STEPS; ++s) {
        const float w = p[s] * inv;
        const float4* v4 = (const float4*)(base + s * 384 + 256 + h * C_DIM);
#pragma unroll
        for (int i = 0; i < 8; ++i) {
            float4 vv = v4[i];
            out[i].x += w * vv.x; out[i].y += w * vv.y;
            out[i].z += w * vv.z; out[i].w += w * vv.w;
        }
    }
    float4* op = (float4*)(o + ((size_t)n * T_STEPS + t) * F_DIM + h * C_DIM);
#pragma unroll
    for (int i = 0; i < 8; ++i) op[i] = out[i];
}

// ---------------------------------------------------------------------------
// LayerNorm kernels: one wave per 128-float row, lane owns 4 consecutive cols.
// ---------------------------------------------------------------------------
__device__ inline float wave_sum(float v) {
#pragma unroll
    for (int off = 16; off > 0; off >>= 1) v += __shfl_xor(v, off, 32);
    return v;
}

__global__ __launch_bounds__(256)
void ln1_kernel(const float* __restrict__ x, const float* __restrict__ o,
                const float* __restrict__ g, const float* __restrict__ b,
                float* __restrict__ y)
{
    int row = blockIdx.x * (blockDim.x >> 5) + (threadIdx.x >> 5);
    if (row >= M_ROWS) return;
    int lane = threadIdx.x & 31;
    float4 xv = ((const float4*)(x + (size_t)row * F_DIM))[lane];
    float4 ov = ((const float4*)(o + (size_t)row * F_DIM))[lane];
    float4 s; s.x = xv.x + ov.x; s.y = xv.y + ov.y; s.z = xv.z + ov.z; s.w = xv.w + ov.w;
    float mean = wave_sum(s.x + s.y + s.z + s.w) * (1.f / F_DIM);
    float dx = s.x - mean, dy = s.y - mean, dz = s.z - mean, dw = s.w - mean;
    float var = wave_sum(dx * dx + dy * dy + dz * dz + dw * dw) * (1.f / F_DIM);
    float rstd = rsqrtf(var + 1e-5f);
    float4 gv = ((const float4*)g)[lane];
    float4 bv = ((const float4*)b)[lane];
    float4 r;
    r.x = dx * rstd * gv.x + bv.x; r.y = dy * rstd * gv.y + bv.y;
    r.z = dz * rstd * gv.z + bv.z; r.w = dw * rstd * gv.w + bv.w;
    ((float4*)(y + (size_t)row * F_DIM))[lane] = r;
}

__global__ __launch_bounds__(256)
void ln2_kernel(const float* __restrict__ x1, const float* __restrict__ agg,
                const float* __restrict__ den, const float* __restrict__ skip,
                const float* __restrict__ g, const float* __restrict__ b,
                float* __restrict__ y)
{
    int row = blockIdx.x * (blockDim.x >> 5) + (threadIdx.x >> 5);
    if (row >= M_ROWS) return;
    int lane = threadIdx.x & 31;
    int h = lane >> 3;                          // 4 floats per lane, 32 per head
    float d = den[(size_t)row * H_HEADS + h] + 1e-16f;
    float invd = 1.f / d;
    float4 xv = ((const float4*)(x1 + (size_t)row * F_DIM))[lane];
    float4 av = ((const float4*)(agg + (size_t)row * F_DIM))[lane];
    float4 kv = ((const float4*)(skip + (size_t)row * F_DIM))[lane];
    float4 s;
    s.x = xv.x + av.x * invd + kv.x; s.y = xv.y + av.y * invd + kv.y;
    s.z = xv.z + av.z * invd + kv.z; s.w = xv.w + av.w * invd + kv.w;
    float mean = wave_sum(s.x + s.y + s.z + s.w) * (1.f / F_DIM);
    float dx = s.x - mean, dy = s.y - mean, dz = s.z - mean, dw = s.w - mean;
    float var = wave_sum(dx * dx + dy * dy + dz * dz + dw * dw) * (1.f / F_DIM);
    float rstd = rsqrtf(var + 1e-5f);
    float4 gv = ((const float4*)g)[lane];
    float4 bv = ((const float4*)b)[lane];
    float4 r;
    r.x = dx * rstd * gv.x + bv.x; r.y = dy * rstd * gv.y + bv.y;
    r.z = dz * rstd * gv.z + bv.z; r.w = dw * rstd * gv.w + bv.w;
    ((float4*)(y + (size_t)row * F_DIM))[lane] = r;
}

// ---------------------------------------------------------------------------
// Edge (spatial TransformerConv) kernels.
// ---------------------------------------------------------------------------
__global__ __launch_bounds__(256)
void init_spatial(float* __restrict__ mx, float* __restrict__ den,
                  float* __restrict__ agg)
{
    size_t i = (size_t)blockIdx.x * blockDim.x + threadIdx.x;
    const size_t nAgg = (size_t)M_ROWS * F_DIM;
    const size_t nMx = (size_t)M_ROWS * H_HEADS;
    if (i < nAgg) agg[i] = 0.f;
    if (i < nMx) { mx[i] = -INFINITY; den[i] = 0.f; }
}

__device__ inline void atomicMaxF(float* addr, float val) {
    if (val >= 0.f) atomicMax((int*)addr, __float_as_int(val));
    else            atomicMin((unsigned int*)addr, __float_as_uint(val));
}

// thread per (edge, t): 4-head dot products sq[dst] . sk[src]
__global__ __launch_bounds__(256)
void edge_logits(const long long* __restrict__ ei, const float* __restrict__ sq,
                 const float* __restrict__ sk, float* __restrict__ logits,
                 float* __restrict__ mx)
{
    size_t idx = (size_t)blockIdx.x * blockDim.x + threadIdx.x;
    if (idx >= (size_t)E_EDGES * T_STEPS) return;
    const int t = (int)(idx % T_STEPS);
    const size_t e = idx / T_STEPS;
    const long long src = ei[e];
    const long long dst = ei[E_EDGES + e];
    const float4* qd = (const float4*)(sq + ((size_t)dst * T_STEPS + t) * F_DIM);
    const float4* ks = (const float4*)(sk + ((size_t)src * T_STEPS + t) * F_DIM);
    float acc[H_HEADS] = {0.f, 0.f, 0.f, 0.f};
#pragma unroll
    for (int i = 0; i < 32; ++i) {
        float4 a = qd[i], b = ks[i];
        acc[i >> 3] += a.x * b.x + a.y * b.y + a.z * b.z + a.w * b.w;
    }
    float* mrow = mx + ((size_t)dst * T_STEPS + t) * H_HEADS;
#pragma unroll
    for (int h = 0; h < H_HEADS; ++h) {
        float lg = acc[h] * SCALE;
        logits[idx * H_HEADS + h] = lg;
        atomicMaxF(&mrow[h], lg);
    }
}

// thread per (edge, t): e = exp(logit - mx[dst]); scatter den and e*sv[src].
__global__ __launch_bounds__(256)
void edge_scatter(const long long* __restrict__ ei, const float* __restrict__ sv,
                  const float* __restrict__ logits, const float* __restrict__ mx,
                  float* __restrict__ den, float* __restrict__ agg)
{
    size_t idx = (size_t)blockIdx.x * blockDim.x + threadIdx.x;
    if (idx >= (size_t)E_EDGES * T_STEPS) return;
    const int t = (int)(idx % T_STEPS);
    const size_t e = idx / T_STEPS;
    const long long src = ei[e];
    const long long dst = ei[E_EDGES + e];
    const float* mrow = mx + ((size_t)dst * T_STEPS + t) * H_HEADS;
    float* drow = den + ((size_t)dst * T_STEPS + t) * H_HEADS;
    float w[H_HEADS];
#pragma unroll
    for (int h = 0; h < H_HEADS; ++h) {
        w[h] = __expf(logits[idx * H_HEADS + h] - mrow[h]);
        atomicAdd(&drow[h], w[h]);
    }
    const float4* svr = (const float4*)(sv + ((size_t)src * T_STEPS + t) * F_DIM);
    float* ag = agg + ((size_t)dst * T_STEPS + t) * F_DIM;
#pragma unroll
    for (int i = 0; i < 32; ++i) {
        float4 v = svr[i];
        float ww = w[i >> 3];
        atomicAdd(&ag[4 * i + 0], ww * v.x);
        atomicAdd(&ag[4 * i + 1], ww * v.y);
        atomicAdd(&ag[4 * i + 2], ww * v.z);
        atomicAdd(&ag[4 * i + 3], ww * v.w);
    }
}

// ---------------------------------------------------------------------------
// Host-side orchestration.
// ---------------------------------------------------------------------------
static inline void launch_gemm(const float* A, const float* W, const float* bias,
                               const float* resid, float* Y, int Mrows, int Nout,
                               int K, int relu, hipStream_t stream)
{
    int totalWaves = (Mrows >> 4) * (Nout >> 6);
    int blocks = (totalWaves + 7) / 8;          // 8 waves per 256-thread block
    gemm_wmma_f32<<<blocks, 256, 0, stream>>>(A, W, bias, resid, Y, Mrows, Nout, K, relu);
}

extern "C" void kernel_launch(void* const* d_in, const int* in_sizes, int n_in,
                              void* d_out, int out_size, void* d_ws, size_t ws_size,
                              hipStream_t stream)
{
    (void)in_sizes; (void)n_in; (void)out_size; (void)ws_size;
    const float* x          = (const float*)d_in[0];
    const long long* eidx   = (const long long*)d_in[1];     // int64 edge index
    const float* in_proj_w  = (const float*)d_in[2];
    const float* in_proj_b  = (const float*)d_in[3];
    const float* out_proj_w = (const float*)d_in[4];
    const float* out_proj_b = (const float*)d_in[5];
    const float* Wq = (const float*)d_in[6];  const float* bq = (const float*)d_in[7];
    const float* Wk = (const float*)d_in[8];  const float* bk = (const float*)d_in[9];
    const float* Wv = (const float*)d_in[10]; const float* bv = (const float*)d_in[11];
    const float* Wskip = (const float*)d_in[12]; const float* bskip = (const float*)d_in[13];
    const float* ln1_g = (const float*)d_in[14]; const float* ln1_b = (const float*)d_in[15];
    const float* ln2_g = (const float*)d_in[16]; const float* ln2_b = (const float*)d_in[17];
    const float* Wf1 = (const float*)d_in[18]; const float* bf1 = (const float*)d_in[19];
    const float* Wf2 = (const float*)d_in[20]; const float* bf2 = (const float*)d_in[21];
    float* out = (float*)d_out;

    // workspace layout (floats)
    float* ws = (float*)d_ws;
    size_t off = 0;
    float* qkv = ws + off;      off += (size_t)M_ROWS * 384;   // reused: sq/sk/sv, then ffn1
    float* obuf = ws + off;     off += (size_t)M_ROWS * F_DIM; // attn out, reused: skip
    float* x1 = ws + off;       off += (size_t)M_ROWS * F_DIM;
    float* x2 = ws + off;       off += (size_t)M_ROWS * F_DIM; // oproj first, then x2
    float* logits = ws + off;   off += (size_t)E_EDGES * T_STEPS * H_HEADS;
    float* mx = ws + off;       off += (size_t)M_ROWS * H_HEADS;
    float* den = ws + off;      off += (size_t)M_ROWS * H_HEADS;
    float* agg = ws + off;      off += (size_t)M_ROWS * F_DIM;

    float* sq = qkv;
    float* sk = qkv + (size_t)M_ROWS * F_DIM;
    float* sv = qkv + (size_t)2 * M_ROWS * F_DIM;
    float* skip = obuf;
    float* ffn1 = qkv;
    float* oproj = x2;

    // 1) temporal QKV projection  [M,128] @ [384,128]^T -> [M,384]
    launch_gemm(x, in_proj_w, in_proj_b, nullptr, qkv, M_ROWS, 384, F_DIM, 0, stream);

    // 2) temporal attention -> obuf [M,128]
    temporal_attn<<<(N_NODES * H_HEADS * T_STEPS + 255) / 256, 256, 0, stream>>>(qkv, obuf);

    // 3) output projection -> oproj
    launch_gemm(obuf, out_proj_w, out_proj_b, nullptr, oproj, M_ROWS, F_DIM, F_DIM, 0, stream);

    // 4) x1 = LN1(x + oproj)
    ln1_kernel<<<(M_ROWS + 7) / 8, 256, 0, stream>>>(x, oproj, ln1_g, ln1_b, x1);

    // 5) spatial projections (reuse qkv buffer)
    launch_gemm(x1, Wq, bq, nullptr, sq, M_ROWS, F_DIM, F_DIM, 0, stream);
    launch_gemm(x1, Wk, bk, nullptr, sk, M_ROWS, F_DIM, F_DIM, 0, stream);
    launch_gemm(x1, Wv, bv, nullptr, sv, M_ROWS, F_DIM, F_DIM, 0, stream);
    launch_gemm(x1, Wskip, bskip, nullptr, skip, M_ROWS, F_DIM, F_DIM, 0, stream);

    // 6) segment-softmax init + edge passes
    init_spatial<<<((size_t)M_ROWS * F_DIM + 255) / 256, 256, 0, stream>>>(mx, den, agg);
    {
        size_t nET = (size_t)E_EDGES * T_STEPS;
        edge_logits<<<(nET + 255) / 256, 256, 0, stream>>>(eidx, sq, sk, logits, mx);
        edge_scatter<<<(nET + 255) / 256, 256, 0, stream>>>(eidx, sv, logits, mx, den, agg);
    }

    // 7) x2 = LN2(x1 + agg/den + skip)
    ln2_kernel<<<(M_ROWS + 7) / 8, 256, 0, stream>>>(x1, agg, den, skip, ln2_g, ln2_b, x2);

    // 8) FFN: ffn1 = relu(x2 @ Wf1^T + bf1); out = x2 + ffn1 @ Wf2^T + bf2
    launch_gemm(x2, Wf1, bf1, nullptr, ffn1, M_ROWS, 2 * F_DIM, F_DIM, 1, stream);
    launch_gemm(ffn1, Wf2, bf2, x2, out, M_ROWS, F_DIM, 2 * F_DIM, 0, stream);
}